// Crop_90142773608791
// MI455X (gfx1250) — compile-verified
//
#include <hip/hip_runtime.h>
#include <hip/hip_bf16.h>

// ---- problem constants (from reference) ----
#define B_DIM   64
#define A_DIM   3
#define C_DIM   2
#define F_DIM   199
#define T_DIM   512
#define N_DIM   (B_DIM * A_DIM)        // 192
#define N_ROWS  (C_DIM * F_DIM)        // 398 crop rows per n
#define N_ROWS_PAD 400                 // padded to multiple of 8 for vector stores
#define TL      16                     // T_LEN
#define FRONT   7
#define FC_I    99                     // FC_IDX in the 199-bin band
#define NOISE_LO 39                    // f < 39  (F_ARR < 41000)
#define NOISE_HI 160                   // f >= 160 (F_ARR > 44000)
#define N_NOISE 78
#define WIN_ELEMS (N_ROWS * TL)        // 6368

typedef float         v8f __attribute__((ext_vector_type(8)));
typedef float         v2f __attribute__((ext_vector_type(2)));
typedef unsigned int  v4u __attribute__((ext_vector_type(4)));
typedef int           v8i __attribute__((ext_vector_type(8)));
typedef int           v4i __attribute__((ext_vector_type(4)));

// ---------------------------------------------------------------------------
// Kernel 1: mid[n] = argmax_t ( |x[n,0,FC,t]| + |x[n,1,FC,t]| ), first max.
// ---------------------------------------------------------------------------
__global__ __launch_bounds__(256) void argmax_mid_kernel(
    const float* __restrict__ x, int* __restrict__ midArr) {
  const int n   = blockIdx.x;
  const int tid = threadIdx.x;
  const float* p0 = x + ((size_t)(n * C_DIM + 0) * F_DIM + FC_I) * T_DIM;
  const float* p1 = x + ((size_t)(n * C_DIM + 1) * F_DIM + FC_I) * T_DIM;

  __shared__ float sv[256];
  __shared__ int   si[256];

  float best = -1.0f;
  int   bidx = 0;
  for (int t = tid; t < T_DIM; t += 256) {        // ascending -> first-max tie rule
    float m = fabsf(p0[t]) + fabsf(p1[t]);
    if (m > best) { best = m; bidx = t; }
  }
  sv[tid] = best; si[tid] = bidx;
  __syncthreads();
  for (int s = 128; s > 0; s >>= 1) {
    if (tid < s) {
      if (sv[tid + s] > sv[tid] ||
          (sv[tid + s] == sv[tid] && si[tid + s] < si[tid])) {
        sv[tid] = sv[tid + s]; si[tid] = si[tid + s];
      }
    }
    __syncthreads();
  }
  if (tid == 0) midArr[n] = si[0];
}

// ---------------------------------------------------------------------------
// Kernel 2 (one block per n):
//   TDM : single 2D descriptor DMAs 398 rows x 16 floats (stride 512) to LDS.
//   Pass A: branchless masked crop -> d_out (float4 stores) and |val| -> winAbs.
//   Pass B: pulse row-sums via V_WMMA_F32_16X16X4_F32; A-fragments read
//           unconditionally (ds_load_2addr_b64); D extracted with two
//           unconditional ds_store_b128 into a padded row-sum array.
//   Pass C: noise/sig/SNR -> half_band -> musk.
// ---------------------------------------------------------------------------
__global__ __launch_bounds__(256) void crop_snr_kernel(
    const float* __restrict__ x, const int* __restrict__ midArr,
    float* __restrict__ outCrop, float* __restrict__ outMusk) {

  __shared__ __align__(16) float win[WIN_ELEMS];     // TDM destination
  __shared__ __align__(16) float winAbs[WIN_ELEMS];  // |masked| values for WMMA
  __shared__ __align__(16) float rs[N_ROWS_PAD];     // per-row sums (rows 398,399 = pad)
  __shared__ float pulse[F_DIM];
  __shared__ int   s_hb;

  const int n   = blockIdx.x;
  const int tid = threadIdx.x;

  const int mid   = midArr[n];
  const int s0    = mid - FRONT;                     // first tap index
  const int start = min(max(s0, 0), T_DIM - TL);     // clamped window start
  const int shift = s0 - start;                      // lds col = tap + shift

  // ---- Tensor Data Mover: one 2D tile load (16 x 398, row stride 512) ----
  if (tid < 32) {   // one wave issues the DMA
    unsigned long long gaddr =
        (unsigned long long)(uintptr_t)(x + (size_t)n * (N_ROWS * T_DIM) + start);
    unsigned ldsOff = (unsigned)(uintptr_t)(void*)win;  // flat LDS addr[31:0] = offset

    v4u g0;
    g0.x = 1u;                                          // count=1, user descriptor
    g0.y = ldsOff;                                      // lds_addr
    g0.z = (unsigned)(gaddr & 0xFFFFFFFFu);             // global_addr[31:0]
    g0.w = (unsigned)((gaddr >> 32) & 0x1FFFFFFu)       // global_addr[56:32]
         | (2u << 30);                                  // type = 2 ("image")
    v8i g1;
    g1[0] = (int)(2u << 16);                            // data_size = 4 bytes
    g1[1] = (int)((unsigned)T_DIM << 16);               // tensor_dim0[15:0] = 512
    g1[2] = (int)((unsigned)N_ROWS << 16);              // tensor_dim1[15:0] = 398
    g1[3] = (int)((unsigned)TL << 16);                  // tile_dim0 = 16
    g1[4] = N_ROWS;                                     // tile_dim1 = 398
    g1[5] = T_DIM;                                      // tensor_dim0_stride = 512
    g1[6] = 0;
    g1[7] = 0;
    v4i z4 = {0, 0, 0, 0};
#if defined(__clang_major__) && (__clang_major__ >= 23)
    v8i z8 = {0, 0, 0, 0, 0, 0, 0, 0};
    __builtin_amdgcn_tensor_load_to_lds(g0, g1, z4, z4, z8, 0);
#else
    __builtin_amdgcn_tensor_load_to_lds(g0, g1, z4, z4, 0);
#endif
    __builtin_amdgcn_s_wait_tensorcnt(0);
  }
  __syncthreads();

  // ---- Pass A: branchless masked crop + |val| precompute (float4 I/O) ----
  const size_t cropBase = (size_t)n * WIN_ELEMS;
  for (int e4 = tid * 4; e4 < WIN_ELEMS; e4 += 256 * 4) {
    float4 outv, absv;
#pragma unroll
    for (int q = 0; q < 4; ++q) {
      const int e   = e4 + q;
      const int j   = e & (TL - 1);
      const int idx = s0 + j;                                  // original tap index
      const int src = min(max(e + shift, 0), WIN_ELEMS - 1);   // safe LDS index
      const float v = (idx >= 0 && idx < T_DIM) ? win[src] : 0.0f;
      ((float*)&outv)[q] = v;
      ((float*)&absv)[q] = fabsf(v);
    }
    *(float4*)(outCrop + cropBase + e4) = outv;
    *(float4*)(winAbs + e4)             = absv;
  }
  __syncthreads();   // winAbs ready for WMMA pass

  // ---- Pass B: pulse row-sums via WMMA f32 16x16x4, B = ones ----
  // A layout (ISA 7.12.2, 32-bit A 16x4): lanes 0-15 hold K={0,1}, lanes 16-31 K={2,3}.
  // D layout: lane 0 holds rows rowbase+0..7 (VGPR 0..7); lane 16 rows rowbase+8..15.
  {
    const int lane = tid & 31;
    const int wv   = tid >> 5;
    const int m    = lane & 15;
    const int grp  = lane >> 4;
    const v2f bOnes = {1.0f, 1.0f};
    const int nChunks = (N_ROWS + 15) / 16;             // 25
    for (int chunk = wv; chunk < nChunks; chunk += 8) {
      const int rowbase = chunk * 16;
      const int row     = min(rowbase + m, N_ROWS - 1); // clamp: pad rows unused
      const float* wrow = winAbs + row * TL;            // 64B-aligned row
      v8f acc = {0.f, 0.f, 0.f, 0.f, 0.f, 0.f, 0.f, 0.f};
#pragma unroll
      for (int kc = 0; kc < 4; ++kc) {
        // two adjacent taps per lane -> one 8B-aligned ds_load_b64
        const v2f a = *(const v2f*)(wrow + kc * 4 + grp * 2);
        acc = __builtin_amdgcn_wmma_f32_16x16x4_f32(
            false, a, false, bOnes, (short)0, acc, false, false);
      }
      if (m == 0) {                                     // lanes 0 and 16 extract
        const int rb = rowbase + grp * 8;               // multiple of 8 -> 32B aligned
        float4 lo = {acc[0], acc[1], acc[2], acc[3]};
        float4 hi = {acc[4], acc[5], acc[6], acc[7]};
        *(float4*)(rs + rb)     = lo;                   // unconditional: rs is padded
        *(float4*)(rs + rb + 4) = hi;                   // to 400 entries (rb+7 <= 399)
      }
    }
  }
  __syncthreads();

  if (tid < F_DIM) pulse[tid] = rs[tid] + rs[tid + F_DIM];   // sum over C
  __syncthreads();

  // ---- Pass C: noise / sig / SNR -> half_band ----
  if (tid == 0) {
    float acc = 0.0f;
    for (int f = 0; f < NOISE_LO; ++f)        acc += pulse[f];
    for (int f = NOISE_HI; f < F_DIM; ++f)    acc += pulse[f];
    const float noise = acc / (float)N_NOISE;
    const float sig   = pulse[FC_I];
    const float d     = sig - noise;
    const float snr   = 10.0f * log10f((d * d) / (noise * noise));
    float hbf;
    if      (snr < 30.0f)  hbf = 0.1f * (snr - 28.0f) + 20.0f;
    else if (snr < 40.0f)  hbf = 0.1f * (snr - 33.7f) + 35.0f;
    else if (snr < 46.0f)  hbf = 0.1f * (snr - 42.0f) + 125.0f;
    else if (snr < 47.5f)  hbf = 0.1f * (snr - 46.5f) + 300.0f;
    else                   hbf = 0.1f * (snr - 47.5f) + 500.0f;
    s_hb = (int)hbf;                                  // trunc toward zero, as astype(int32)
  }
  __syncthreads();

  // ---- musk: (N, F, 16) float band mask ----
  const int hb = s_hb;
  const size_t muskBase = (size_t)n * (F_DIM * TL);
  for (int e4 = tid * 4; e4 < F_DIM * TL; e4 += 256 * 4) {
    const int f = e4 >> 4;                              // 4 elems share one f
    const float v = (f >= FC_I - hb && f < FC_I + hb) ? 1.0f : 0.0f;
    float4 o = {v, v, v, v};
    *(float4*)(outMusk + muskBase + e4) = o;
  }
}

// ---------------------------------------------------------------------------
extern "C" void kernel_launch(void* const* d_in, const int* in_sizes, int n_in,
                              void* d_out, int out_size, void* d_ws, size_t ws_size,
                              hipStream_t stream) {
  const float* x = (const float*)d_in[0];
  int*   midArr  = (int*)d_ws;                        // 192 ints of scratch
  float* outCrop = (float*)d_out;                     // (64,3,2,199,16) flat
  float* outMusk = outCrop + (size_t)N_DIM * WIN_ELEMS;    // (64,3,199,16) flat

  argmax_mid_kernel<<<N_DIM, 256, 0, stream>>>(x, midArr);
  crop_snr_kernel<<<N_DIM, 256, 0, stream>>>(x, midArr, outCrop, outMusk);
}